// NormalMM_80152679678262
// MI455X (gfx1250) — compile-verified
//
#include <hip/hip_runtime.h>
#include <math.h>

#define NUM_STATES  32
#define NUM_TYPES   128
#define NUM_GENES   8192
#define NUM_SAMPLES 8192
#define LOG_2PI     1.8378770664093453f

typedef float v2f __attribute__((ext_vector_type(2)));
typedef float v4f __attribute__((ext_vector_type(4)));
typedef float v8f __attribute__((ext_vector_type(8)));

// ---------------------------------------------------------------------------
// Kernel 1: WMMA GEMMs over the tiny (32x128) x (128x8192) problem.
// Computes per (state, gene):
//   mu    = cp @ Zmu
//   sigma = cp^2 @ Zsig + (cp @ Zsig)^2 - cp^2 @ Zsig^2
// and stores mu, 1/sigma, log(sigma) tables so the big elementwise pass
// needs no divides / transcendentals.
// One wave per 16x16 output tile: 2 M-tiles x 512 N-tiles = 1024 waves.
// ---------------------------------------------------------------------------
__global__ __launch_bounds__(256) void build_tables_kernel(
    const float* __restrict__ cp,    // (32, 128)
    const float* __restrict__ Zmu,   // (128, 8192)
    const float* __restrict__ Zsig,  // (128, 8192)
    float* __restrict__ muT,         // (32, 8192)
    float* __restrict__ invT,        // (32, 8192)
    float* __restrict__ logT)        // (32, 8192)
{
    const int wave  = (blockIdx.x * blockDim.x + threadIdx.x) >> 5;
    const int lane  = threadIdx.x & 31;
    const int nTile = wave & 511;       // 512 N tiles
    const int mTile = wave >> 9;        // 2 M tiles
    const int m0 = mTile * 16;
    const int n0 = nTile * 16;
    const int nl = lane & 15;           // row (A) / col (B,C) index within tile
    const int kh = lane >> 4;           // K half-select per ISA f32 layouts

    v8f accMu = {}, accS1 = {}, accS0 = {}, accS2 = {};

    const int arow = m0 + nl;           // A-matrix row for this lane
    for (int k0 = 0; k0 < NUM_TYPES; k0 += 4) {
        const int ka = k0 + 2 * kh;     // this lane-half covers K = ka, ka+1

        // A fragment (16x4 f32): VGPR0 = K even-of-half, VGPR1 = K odd-of-half
        float a0 = cp[arow * NUM_TYPES + ka];
        float a1 = cp[arow * NUM_TYPES + ka + 1];
        v2f aCp;  aCp.x  = a0;      aCp.y  = a1;
        v2f aCp2; aCp2.x = a0 * a0; aCp2.y = a1 * a1;

        // B fragments (4x16 f32): row ka in VGPR0, row ka+1 in VGPR1, N = lane&15
        size_t bi0 = (size_t)ka * NUM_GENES + n0 + nl;
        size_t bi1 = (size_t)(ka + 1) * NUM_GENES + n0 + nl;
        float bm0 = Zmu[bi0],  bm1 = Zmu[bi1];
        float bs0 = Zsig[bi0], bs1 = Zsig[bi1];
        v2f bMu;  bMu.x  = bm0;       bMu.y  = bm1;
        v2f bSg;  bSg.x  = bs0;       bSg.y  = bs1;
        v2f bSg2; bSg2.x = bs0 * bs0; bSg2.y = bs1 * bs1;

        accMu = __builtin_amdgcn_wmma_f32_16x16x4_f32(false, aCp,  false, bMu,
                                                      (short)0, accMu, false, false);
        accS1 = __builtin_amdgcn_wmma_f32_16x16x4_f32(false, aCp,  false, bSg,
                                                      (short)0, accS1, false, false);
        accS0 = __builtin_amdgcn_wmma_f32_16x16x4_f32(false, aCp2, false, bSg,
                                                      (short)0, accS0, false, false);
        accS2 = __builtin_amdgcn_wmma_f32_16x16x4_f32(false, aCp2, false, bSg2,
                                                      (short)0, accS2, false, false);
    }

    // C/D layout: VGPR r, lanes 0-15 -> M=r, lanes 16-31 -> M=r+8; N = lane&15
#pragma unroll
    for (int r = 0; r < 8; ++r) {
        int m = m0 + r + 8 * kh;
        size_t idx = (size_t)m * NUM_GENES + n0 + nl;
        float sg = accS0[r] + accS1[r] * accS1[r] - accS2[r];
        muT[idx]  = accMu[r];
        invT[idx] = 1.0f / sg;
        logT[idx] = __logf(sg);
    }
}

// ---------------------------------------------------------------------------
// Kernel 2: bandwidth-bound elementwise pass. One block per sample row.
// logp = -0.5*((x-mu)*inv)^2 - logsg - 0.5*log(2pi)
// X read and logp write are pure streams (no reuse) -> non-temporal (TH=NT)
// so they don't evict the 3 MB state tables, which are reused by all 8192
// rows and should stay resident in L2 (regular-temporal loads).
// Deterministic per-row reduction -> partials[row].
// ---------------------------------------------------------------------------
__global__ __launch_bounds__(256) void logp_kernel(
    const float* __restrict__ X,       // (8192, 8192)
    const int*   __restrict__ day,     // (8192,)
    const float* __restrict__ muT,
    const float* __restrict__ invT,
    const float* __restrict__ logT,
    float* __restrict__ logp,          // (8192, 8192) -> d_out + 1
    float* __restrict__ partials)      // (8192,)
{
    __shared__ float sdata[256];
    const int row   = blockIdx.x;
    const int state = day[row] - 1;

    const v4f* xr  = (const v4f*)(X    + (size_t)row   * NUM_GENES);
    const v4f* mr  = (const v4f*)(muT  + (size_t)state * NUM_GENES);
    const v4f* ir  = (const v4f*)(invT + (size_t)state * NUM_GENES);
    const v4f* lr  = (const v4f*)(logT + (size_t)state * NUM_GENES);
    v4f*       oro = (v4f*)(logp + (size_t)row * NUM_GENES);

    const float c = -0.5f * LOG_2PI;
    float lsum = 0.0f;

    for (int i = threadIdx.x; i < NUM_GENES / 4; i += 256) {
        v4f x  = __builtin_nontemporal_load(xr + i);   // stream, no reuse
        v4f mu = mr[i];                                // hot table, keep cached
        v4f iv = ir[i];
        v4f lg = lr[i];
        v4f o;
        float z;
        z = (x.x - mu.x) * iv.x; o.x = fmaf(-0.5f * z, z, c) - lg.x;
        z = (x.y - mu.y) * iv.y; o.y = fmaf(-0.5f * z, z, c) - lg.y;
        z = (x.z - mu.z) * iv.z; o.z = fmaf(-0.5f * z, z, c) - lg.z;
        z = (x.w - mu.w) * iv.w; o.w = fmaf(-0.5f * z, z, c) - lg.w;
        __builtin_nontemporal_store(o, oro + i);       // stream, write-once
        lsum += (o.x + o.y) + (o.z + o.w);
    }

    sdata[threadIdx.x] = lsum;
    __syncthreads();
#pragma unroll
    for (int s = 128; s > 0; s >>= 1) {
        if (threadIdx.x < (unsigned)s) sdata[threadIdx.x] += sdata[threadIdx.x + s];
        __syncthreads();
    }
    if (threadIdx.x == 0) partials[row] = sdata[0];
}

// ---------------------------------------------------------------------------
// Kernel 3: deterministic final reduction of 8192 row partials -> d_out[0].
// ---------------------------------------------------------------------------
__global__ __launch_bounds__(256) void reduce_final_kernel(
    const float* __restrict__ partials, float* __restrict__ out0)
{
    __shared__ float sdata[256];
    float s = 0.0f;
    for (int i = threadIdx.x; i < NUM_SAMPLES; i += 256) s += partials[i];
    sdata[threadIdx.x] = s;
    __syncthreads();
#pragma unroll
    for (int st = 128; st > 0; st >>= 1) {
        if (threadIdx.x < (unsigned)st) sdata[threadIdx.x] += sdata[threadIdx.x + st];
        __syncthreads();
    }
    if (threadIdx.x == 0) out0[0] = sdata[0];
}

extern "C" void kernel_launch(void* const* d_in, const int* in_sizes, int n_in,
                              void* d_out, int out_size, void* d_ws, size_t ws_size,
                              hipStream_t stream) {
    const float* cp   = (const float*)d_in[0];  // cell_prob (32,128)
    const float* Zmu  = (const float*)d_in[1];  // (128,8192)
    const float* Zsig = (const float*)d_in[2];  // (128,8192)
    const float* X    = (const float*)d_in[3];  // (8192,8192)
    const int*   day  = (const int*)d_in[4];    // (8192,)

    float* out = (float*)d_out;                 // [0]=sum, [1..]=logp
    float* ws  = (float*)d_ws;

    const size_t tab = (size_t)NUM_STATES * NUM_GENES;   // 262144 floats
    float* muT      = ws;
    float* invT     = ws + tab;
    float* logT     = ws + 2 * tab;
    float* partials = ws + 3 * tab;

    // 1024 waves = 1024 16x16 tiles (2 M-tiles x 512 N-tiles)
    build_tables_kernel<<<128, 256, 0, stream>>>(cp, Zmu, Zsig, muT, invT, logT);
    logp_kernel<<<NUM_SAMPLES, 256, 0, stream>>>(X, day, muT, invT, logT,
                                                 out + 1, partials);
    reduce_final_kernel<<<1, 256, 0, stream>>>(partials, out);
}